// ECTLoss_84490596647649
// MI455X (gfx1250) — compile-verified
//
#include <hip/hip_runtime.h>

// ECT (Euler Characteristic Transform) for MI455X / gfx1250, wave32.
// Shapes from the harness: V=131072, B=2, C=4 (BC=8), D=64 dirs, R=64 bins.
//
// Pipeline:
//   1) memsetAsync: zero global hist[d][r][bc] (64*64*8 f32 = 128KB) in d_ws
//   2) ect_hist_kernel: per block = 1024 voxels x 16 directions
//        - in-block softmax -> LDS wt[1024][8]
//        - heights via V_WMMA_F32_16X16X4_F32 (K=3 padded to 4)
//        - ds_add_f32 scatter into padded LDS hist[16][64][8]
//        - global_atomic_add_f32 merge into d_ws hist
//   3) ect_cumsum_kernel: prefix-sum over R + transpose to [B,C,D,R]

typedef float v2f __attribute__((ext_vector_type(2)));
typedef float v8f __attribute__((ext_vector_type(8)));

#define THREADS 256
#define WAVES   (THREADS / 32)
#define VPB     1024          // voxels per block (divides V=131072)
#define DPB     16            // directions per block = one WMMA N tile
#define TPW     (VPB / 16 / WAVES)   // tiles per wave = 8 (compile-time)
#define RES     64            // resolution (bins)
#define BCN     8             // B*C channels
#define HPAD    4             // 516-dword row stride -> lanes spread over banks
#define HSTRIDE (RES * BCN + HPAD)

__global__ __launch_bounds__(THREADS)
void ect_hist_kernel(const float* __restrict__ coords,   // [V,3]
                     const float* __restrict__ logits,   // [B,C,V] = [8][V] flat
                     const float* __restrict__ dirs,     // [3,D]
                     const float* __restrict__ radius_p, // [1]
                     float* __restrict__ g_hist,         // [D][R][BCN]
                     int V, int D)
{
    __shared__ float s_wt[VPB * BCN];          // softmax weights [v_local][bc]
    __shared__ float s_hist[DPB * HSTRIDE];    // padded [n][r*8+bc]

    const int tid    = threadIdx.x;
    const int lane   = tid & 31;
    const int wave   = tid >> 5;
    const int vChunk = blockIdx.x * VPB;
    const int dBase  = blockIdx.y * DPB;

    const float radius = radius_p[0];
    const float safeR  = (radius > 0.0f) ? radius : 1.0f;
    const float scale  = (float)(RES - 1) / (2.0f * safeR);
    const float bias   = safeR * scale;        // bin = rint(h*scale + bias)

    // ---- zero LDS histogram ----
    for (int j = tid; j < DPB * HSTRIDE; j += THREADS) s_hist[j] = 0.0f;

    // ---- softmax over C=4 for both batches, into LDS ----
    for (int i = tid; i < VPB; i += THREADS) {
        const int v = vChunk + i;
        #pragma unroll
        for (int b = 0; b < 2; ++b) {
            float l0 = logits[(b * 4 + 0) * V + v];
            float l1 = logits[(b * 4 + 1) * V + v];
            float l2 = logits[(b * 4 + 2) * V + v];
            float l3 = logits[(b * 4 + 3) * V + v];
            float mx = fmaxf(fmaxf(l0, l1), fmaxf(l2, l3));
            float e0 = __expf(l0 - mx);
            float e1 = __expf(l1 - mx);
            float e2 = __expf(l2 - mx);
            float e3 = __expf(l3 - mx);
            float inv = 1.0f / (e0 + e1 + e2 + e3);
            s_wt[i * BCN + b * 4 + 0] = e0 * inv;
            s_wt[i * BCN + b * 4 + 1] = e1 * inv;
            s_wt[i * BCN + b * 4 + 2] = e2 * inv;
            s_wt[i * BCN + b * 4 + 3] = e3 * inv;
        }
    }
    __syncthreads();

    // ---- B matrix (4x16 direction tile), constant per block ----
    // ISA layout (mirrors A 16x4): lanes 0-15 hold {K0,K1}={x,y},
    // lanes 16-31 hold {K2,K3}={z,0}; N = lane & 15.
    const int  n    = lane & 15;
    const bool hi   = (lane >= 16);
    const int  mOff = hi ? 8 : 0;
    v2f bmat;
    bmat.x = dirs[(hi ? 2 : 0) * D + dBase + n];        // x (lo) / z (hi)
    bmat.y = hi ? 0.0f : dirs[1 * D + dBase + n];       // y (lo) / pad (hi)

    // ---- tile loop: uniform trip count -> no exec masking around WMMA ----
    #pragma unroll 2
    for (int i = 0; i < TPW; ++i) {
        const int t     = wave + i * WAVES;
        const int vTile = vChunk + t * 16;
        const int base  = 3 * (vTile + n);

        // A matrix (16 voxels x 4), branch-free:
        // lanes 0-15 = {x,y}, lanes 16-31 = {z,0}
        v2f a;
        a.x = coords[base + (hi ? 2 : 0)];
        {
            float cy = coords[base + 1];                // always in-bounds
            a.y = hi ? 0.0f : cy;
        }

        v8f c = {};
        c = __builtin_amdgcn_wmma_f32_16x16x4_f32(
                /*neg_a=*/false, a, /*neg_b=*/false, bmat,
                /*c_mod=*/(short)0, c, /*reuse_a=*/false, /*reuse_b=*/false);

        // D layout: VGPR r -> row M=r (lanes 0-15) or M=8+r (lanes 16-31)
        #pragma unroll
        for (int r = 0; r < 8; ++r) {
            int bin = (int)rintf(fmaf(c[r], scale, bias));   // round-half-even
            bin = min(max(bin, 0), RES - 1);
            const float4* w = (const float4*)&s_wt[(t * 16 + mOff + r) * BCN];
            float4 w0 = w[0];                                // ds_load_b128
            float4 w1 = w[1];
            float* hrow = &s_hist[n * HSTRIDE + bin * BCN];
            atomicAdd(&hrow[0], w0.x);                       // ds_add_f32 x8
            atomicAdd(&hrow[1], w0.y);
            atomicAdd(&hrow[2], w0.z);
            atomicAdd(&hrow[3], w0.w);
            atomicAdd(&hrow[4], w1.x);
            atomicAdd(&hrow[5], w1.y);
            atomicAdd(&hrow[6], w1.z);
            atomicAdd(&hrow[7], w1.w);
        }
    }
    __syncthreads();

    // ---- merge block-private histogram into global ----
    for (int j = tid; j < DPB * RES * BCN; j += THREADS) {
        const int dn = j / (RES * BCN);
        const int rb = j - dn * (RES * BCN);
        unsafeAtomicAdd(&g_hist[(dBase + dn) * (RES * BCN) + rb],
                        s_hist[dn * HSTRIDE + rb]);      // global_atomic_add_f32
    }
}

// cumsum over R and transpose [D][R][BC] -> [B][C][D][R] (bc-major)
__global__ void ect_cumsum_kernel(const float* __restrict__ g_hist,
                                  float* __restrict__ out, int D)
{
    const int t = blockIdx.x * blockDim.x + threadIdx.x;
    if (t >= D * BCN) return;
    const int bc = t & (BCN - 1);
    const int d  = t >> 3;
    float acc = 0.0f;
    for (int r = 0; r < RES; ++r) {
        acc += g_hist[(d * RES + r) * BCN + bc];
        out[(bc * D + d) * RES + r] = acc;   // [b][c][d][r] with bc = b*4+c
    }
}

extern "C" void kernel_launch(void* const* d_in, const int* in_sizes, int n_in,
                              void* d_out, int out_size, void* d_ws, size_t ws_size,
                              hipStream_t stream)
{
    const float* coords  = (const float*)d_in[0];   // [V,3]
    const float* logits  = (const float*)d_in[1];   // [2,4,V]
    const float* dirs    = (const float*)d_in[2];   // [3,D]
    const float* radius  = (const float*)d_in[3];   // [1]
    (void)n_in; (void)ws_size; (void)out_size;

    const int V = in_sizes[0] / 3;     // 131072
    const int D = in_sizes[2] / 3;     // 64

    float* g_hist = (float*)d_ws;      // [D][R][BCN] = 128 KB

    hipMemsetAsync(g_hist, 0, (size_t)D * RES * BCN * sizeof(float), stream);

    dim3 grid(V / VPB, D / DPB);       // 128 x 4 blocks
    ect_hist_kernel<<<grid, THREADS, 0, stream>>>(coords, logits, dirs, radius,
                                                  g_hist, V, D);

    const int nseq = D * BCN;          // 512
    ect_cumsum_kernel<<<(nseq + 255) / 256, 256, 0, stream>>>(g_hist,
                                                              (float*)d_out, D);
}